// BaseMoe_54494545051820
// MI455X (gfx1250) — compile-verified
//
#include <hip/hip_runtime.h>

#define TOKENS 16384
#define H_DIM  2048
#define I_DIM  1024
#define NEXP   8
#define TILE_M 32
#define KC     64
#define XPITCH 68   // 68 floats = 272B: 16B-aligned rows, conflict-free A reads

typedef float v2f __attribute__((ext_vector_type(2)));
typedef float v4f __attribute__((ext_vector_type(4)));
typedef float v8f __attribute__((ext_vector_type(8)));

__device__ __forceinline__ float wave_sum(float v) {
#pragma unroll
    for (int off = 16; off > 0; off >>= 1)
        v += __shfl_xor(v, off, 32);
    return v;
}

// ---------------------------------------------------------------------------
// Stage 1: router. One 256-thread block (8 wave32) per token.
// ---------------------------------------------------------------------------
__global__ __launch_bounds__(256) void moe_router_kernel(
    const float* __restrict__ x, const float* __restrict__ gate_w,
    float* __restrict__ logits_out, int* __restrict__ cnt,
    int* __restrict__ lidx, float* __restrict__ lw)
{
    __shared__ float xs[H_DIM];
    __shared__ float lg[NEXP];
    const int t   = blockIdx.x;
    const int tid = threadIdx.x;

    const float* xr = x + (size_t)t * H_DIM;
    for (int i = tid; i < H_DIM; i += 256) xs[i] = xr[i];
    __syncthreads();

    const int wave = tid >> 5, lane = tid & 31;
    {
        const float* gw = gate_w + wave * H_DIM;   // exactly 8 waves = 8 experts
        float s = 0.0f;
        for (int i = lane; i < H_DIM; i += 32) s += xs[i] * gw[i];
        s = wave_sum(s);
        if (lane == 0) lg[wave] = s;
    }
    __syncthreads();

    if (tid < NEXP) logits_out[(size_t)t * NEXP + tid] = lg[tid];

    if (tid == 0) {
        float p[NEXP];
        float mx = lg[0];
#pragma unroll
        for (int e = 1; e < NEXP; ++e) mx = fmaxf(mx, lg[e]);
        float den = 0.0f;
#pragma unroll
        for (int e = 0; e < NEXP; ++e) { p[e] = __expf(lg[e] - mx); den += p[e]; }
        const float inv = 1.0f / den;
#pragma unroll
        for (int e = 0; e < NEXP; ++e) p[e] *= inv;

        int i0 = 0;
#pragma unroll
        for (int e = 1; e < NEXP; ++e) if (p[e] > p[i0]) i0 = e;
        int i1 = (i0 == 0) ? 1 : 0;
#pragma unroll
        for (int e = 0; e < NEXP; ++e) if (e != i0 && p[e] > p[i1]) i1 = e;

        int p0 = atomicAdd(&cnt[i0], 1);
        lidx[i0 * TOKENS + p0] = t;  lw[i0 * TOKENS + p0] = p[i0];
        int p1 = atomicAdd(&cnt[i1], 1);
        lidx[i1 * TOKENS + p1] = t;  lw[i1 * TOKENS + p1] = p[i1];
    }
}

// ---------------------------------------------------------------------------
// Stage 2: fused per-expert SwiGLU MLP on dispatched tokens.
// 512 threads = 16 wave32; tile = 32 gathered token rows (2 WMMA M-subtiles,
// B loaded once per pair of WMMAs). X is staged through LDS with a
// register-prefetch double buffer (b128 global load + b128 LDS store per
// thread per K-chunk) so the next chunk's latency hides under 256 WMMAs.
// ---------------------------------------------------------------------------
__global__ __launch_bounds__(512) void moe_expert_kernel(
    const float* __restrict__ x,
    const float* __restrict__ wg, const float* __restrict__ wu,
    const float* __restrict__ wd,
    const int* __restrict__ cnt, const int* __restrict__ lidx,
    const float* __restrict__ lw, float* __restrict__ out)
{
    __shared__ __align__(16) float sX[TILE_M][XPITCH];
    __shared__ float sHd[TILE_M][I_DIM + 1];      // +1 pad
    __shared__ int   sTok[TILE_M];
    __shared__ float sC[TILE_M];

    const int bx   = blockIdx.x;
    const int e    = bx >> 9;                     // 512 tiles per expert
    const int tile = bx & 511;
    const int n_e  = cnt[e];
    if (tile * TILE_M >= n_e) return;             // block-uniform exit

    const int tid  = threadIdx.x;
    const int wave = tid >> 5, lane = tid & 31;
    const int half = lane >> 4, l16 = lane & 15;

    if (tid < TILE_M) {
        int p  = tile * TILE_M + tid;
        int pc = (p < n_e) ? p : (n_e - 1);       // clamp for gather; masked later
        sTok[tid] = lidx[e * TOKENS + pc];
        sC[tid]   = (p < n_e) ? lw[e * TOKENS + pc] : 0.0f;
    }
    __syncthreads();

    const float* Wg = wg + (size_t)e * H_DIM * I_DIM;
    const float* Wu = wu + (size_t)e * H_DIM * I_DIM;
    const float* Wd = wd + (size_t)e * I_DIM * H_DIM;

    // Staging geometry: thread -> (row, 4-col group); fixed across chunks.
    const int srow = tid >> 4;                    // 32 rows
    const int sc4  = (tid & 15) << 2;             // 16 groups of 4 cols
    const float* xrow = x + (size_t)sTok[srow] * H_DIM + sc4;

    // ---------------- Phase 1: G / U ----------------
    v8f accG[2][4], accU[2][4];
    {
        v8f z = {};
#pragma unroll
        for (int st = 0; st < 2; ++st)
#pragma unroll
            for (int i = 0; i < 4; ++i) { accG[st][i] = z; accU[st][i] = z; }
    }
    const int ncol0 = wave * 64;                  // 16 waves cover I_DIM = 1024

    v4f rx = *(const v4f*)xrow;                   // prefetch chunk 0

    for (int kc0 = 0; kc0 < H_DIM; kc0 += KC) {
        __syncthreads();                          // previous chunk's readers done
        *(v4f*)&sX[srow][sc4] = rx;               // publish current chunk
        __syncthreads();
        if (kc0 + KC < H_DIM)                     // prefetch next chunk
            rx = *(const v4f*)(xrow + kc0 + KC);

        for (int k = 0; k < KC; k += 4) {
            v2f a0, a1;                           // A 16x4: lanes0-15 K=0/1, 16-31 K=2/3
            a0.x = sX[l16][k + 2 * half];
            a0.y = sX[l16][k + 2 * half + 1];
            a1.x = sX[16 + l16][k + 2 * half];
            a1.y = sX[16 + l16][k + 2 * half + 1];
            const int kg = kc0 + k + 2 * half;
#pragma unroll
            for (int nt = 0; nt < 4; ++nt) {
                const int n = ncol0 + nt * 16 + l16;
                v2f bg, bu;
                bg.x = Wg[(size_t)kg * I_DIM + n];
                bg.y = Wg[(size_t)(kg + 1) * I_DIM + n];
                bu.x = Wu[(size_t)kg * I_DIM + n];
                bu.y = Wu[(size_t)(kg + 1) * I_DIM + n];
                accG[0][nt] = __builtin_amdgcn_wmma_f32_16x16x4_f32(
                    false, a0, false, bg, (short)0, accG[0][nt], false, false);
                accG[1][nt] = __builtin_amdgcn_wmma_f32_16x16x4_f32(
                    false, a1, false, bg, (short)0, accG[1][nt], false, false);
                accU[0][nt] = __builtin_amdgcn_wmma_f32_16x16x4_f32(
                    false, a0, false, bu, (short)0, accU[0][nt], false, false);
                accU[1][nt] = __builtin_amdgcn_wmma_f32_16x16x4_f32(
                    false, a1, false, bu, (short)0, accU[1][nt], false, false);
            }
        }
    }

    // silu(G) * U -> LDS Hd  (C layout: VGPR i -> M = i + 8*half, N = l16)
#pragma unroll
    for (int st = 0; st < 2; ++st) {
#pragma unroll
        for (int nt = 0; nt < 4; ++nt) {
#pragma unroll
            for (int i = 0; i < 8; ++i) {
                int m = st * 16 + i + 8 * half;
                int n = ncol0 + nt * 16 + l16;
                float g = accG[st][nt][i];
                float u = accU[st][nt][i];
                float sig = __builtin_amdgcn_rcpf(1.0f + __expf(-g));
                sHd[m][n] = g * sig * u;
            }
        }
    }
    __syncthreads();

    // ---------------- Phase 2: Out = Hd * Wd ----------------
    v8f accO[2][8];
    {
        v8f z = {};
#pragma unroll
        for (int st = 0; st < 2; ++st)
#pragma unroll
            for (int i = 0; i < 8; ++i) accO[st][i] = z;
    }
    const int ocol0 = wave * 128;                 // 16 waves cover H_DIM = 2048

    for (int k = 0; k < I_DIM; k += 4) {
        v2f a0, a1;
        a0.x = sHd[l16][k + 2 * half];
        a0.y = sHd[l16][k + 2 * half + 1];
        a1.x = sHd[16 + l16][k + 2 * half];
        a1.y = sHd[16 + l16][k + 2 * half + 1];
        const int kg = k + 2 * half;
#pragma unroll
        for (int nt = 0; nt < 8; ++nt) {
            const int n = ocol0 + nt * 16 + l16;
            v2f b;
            b.x = Wd[(size_t)kg * H_DIM + n];
            b.y = Wd[(size_t)(kg + 1) * H_DIM + n];
            accO[0][nt] = __builtin_amdgcn_wmma_f32_16x16x4_f32(
                false, a0, false, b, (short)0, accO[0][nt], false, false);
            accO[1][nt] = __builtin_amdgcn_wmma_f32_16x16x4_f32(
                false, a1, false, b, (short)0, accO[1][nt], false, false);
        }
    }

#pragma unroll
    for (int st = 0; st < 2; ++st) {
#pragma unroll
        for (int nt = 0; nt < 8; ++nt) {
#pragma unroll
            for (int i = 0; i < 8; ++i) {
                int m = st * 16 + i + 8 * half;
                int p = tile * TILE_M + m;
                if (p < n_e) {
                    int n = ocol0 + nt * 16 + l16;
                    atomicAdd(&out[(size_t)sTok[m] * H_DIM + n], accO[st][nt][i] * sC[m]);
                }
            }
        }
    }
}

// ---------------------------------------------------------------------------
extern "C" void kernel_launch(void* const* d_in, const int* in_sizes, int n_in,
                              void* d_out, int out_size, void* d_ws, size_t ws_size,
                              hipStream_t stream)
{
    (void)in_sizes; (void)n_in; (void)out_size; (void)ws_size;
    const float* x      = (const float*)d_in[0];
    const float* gate_w = (const float*)d_in[1];
    const float* wg     = (const float*)d_in[2];
    const float* wu     = (const float*)d_in[3];
    const float* wd     = (const float*)d_in[4];

    float* out        = (float*)d_out;
    float* logits_out = out + (size_t)TOKENS * H_DIM;   // tuple output #2

    char*  ws   = (char*)d_ws;
    int*   cnt  = (int*)ws;                              // 8 counters (64B pad)
    int*   lidx = (int*)(ws + 64);                       // NEXP*TOKENS ints
    float* lw   = (float*)(ws + 64 + sizeof(int) * (size_t)NEXP * TOKENS);

    hipMemsetAsync(cnt, 0, 64, stream);
    hipMemsetAsync(out, 0, sizeof(float) * (size_t)TOKENS * H_DIM, stream);

    moe_router_kernel<<<TOKENS, 256, 0, stream>>>(x, gate_w, logits_out, cnt, lidx, lw);
    moe_expert_kernel<<<NEXP * (TOKENS / TILE_M), 512, 0, stream>>>(
        x, wg, wu, wd, cnt, lidx, lw, out);
}